// AudioGaussianScene_31430570672288
// MI455X (gfx1250) — compile-verified
//
#include <hip/hip_runtime.h>
#include <math.h>

#define NG 2048
#define TD 512
#define FD 256

typedef __attribute__((ext_vector_type(2))) float v2f;
typedef __attribute__((ext_vector_type(8))) float v8f;

// ---------------------------------------------------------------------------
// Prep: per-gaussian constants. c_t = 0.5 / (sigma_t^2 * denom), etc.
// Compacts indices of gaussians with nonzero correlation into corr_list.
// ---------------------------------------------------------------------------
__global__ __launch_bounds__(256) void prep_kernel(
    const float* __restrict__ lst, const float* __restrict__ lsf,
    const float* __restrict__ rrho,
    float* __restrict__ c_t, float* __restrict__ c_f,
    float* __restrict__ ist, float* __restrict__ isf,
    float* __restrict__ rho, float* __restrict__ invden,
    int* __restrict__ corr_list, int* __restrict__ corr_count)
{
    int g = blockIdx.x * blockDim.x + threadIdx.x;
    if (g >= NG) return;
    float st  = expf(lst[g]);
    float sf  = expf(lsf[g]);
    float r   = tanhf(rrho[g]);
    float den = 1.0f - r * r + 1e-6f;
    float id  = 1.0f / den;
    float it  = 1.0f / st;
    float jf  = 1.0f / sf;
    c_t[g] = 0.5f * it * it * id;
    c_f[g] = 0.5f * jf * jf * id;
    ist[g] = it; isf[g] = jf; rho[g] = r; invden[g] = id;
    if (fabsf(r) > 1e-7f) {
        int i = atomicAdd(corr_count, 1);
        corr_list[i] = g;
    }
}

// ---------------------------------------------------------------------------
// Factor matrices. A[t][g] = alpha_g * exp(-c_t (t-mu_t)^2)   (512 x 2048)
//                  B[f][g] =           exp(-c_f (f-mu_f)^2)   (256 x 2048)
// K-contiguous layout feeds the WMMA loads directly.
// ---------------------------------------------------------------------------
__global__ __launch_bounds__(256) void build_A_kernel(
    const float* __restrict__ mu_t, const float* __restrict__ c_t,
    const float* __restrict__ alpha, float* __restrict__ A)
{
    int g = blockIdx.x * blockDim.x + threadIdx.x;
    int t = blockIdx.y;
    float d = (float)t - mu_t[g];
    A[(size_t)t * NG + g] = alpha[g] * expf(-c_t[g] * d * d);
}

__global__ __launch_bounds__(256) void build_B_kernel(
    const float* __restrict__ mu_f, const float* __restrict__ c_f,
    float* __restrict__ B)
{
    int g = blockIdx.x * blockDim.x + threadIdx.x;
    int f = blockIdx.y;
    float d = (float)f - mu_f[g];
    B[(size_t)f * NG + g] = expf(-c_f[g] * d * d);
}

// ---------------------------------------------------------------------------
// out[m][n] = sum_k A[m][k] * B[n][k] via V_WMMA_F32_16X16X4_F32.
// One wave per 16x16 output tile; 512 tiles = 64 blocks x 256 threads exactly
// (EXEC all-1s, as WMMA requires).
// Per-lane operand layout (ISA 7.12.2, 32-bit 16x4 A):
//   lane<16 : row = lane,    K = {k0+0, k0+1}
//   lane>=16: row = lane-16, K = {k0+2, k0+3}
// B (4x16) mirrors with N in place of M.
// ---------------------------------------------------------------------------
__global__ __launch_bounds__(256) void gemm_wmma_kernel(
    const float* __restrict__ A, const float* __restrict__ B,
    float* __restrict__ out)
{
    int wave = (int)((blockIdx.x * blockDim.x + threadIdx.x) >> 5);
    int lane = (int)(threadIdx.x & 31);
    int mt = wave >> 4;                 // 0..31
    int nt = wave & 15;                 // 0..15
    int m_base = mt * 16;
    int n_base = nt * 16;
    int r16   = lane & 15;
    int khalf = (lane >> 4) * 2;        // 0 or 2

    const float* ap = A + (size_t)(m_base + r16) * NG + khalf;
    const float* bp = B + (size_t)(n_base + r16) * NG + khalf;

    v8f acc = {0.f, 0.f, 0.f, 0.f, 0.f, 0.f, 0.f, 0.f};
#pragma unroll 8
    for (int k = 0; k < NG; k += 4) {
        v2f a = *(const v2f*)(ap + k);  // global_load_b64, 8B aligned
        v2f b = *(const v2f*)(bp + k);
        acc = __builtin_amdgcn_wmma_f32_16x16x4_f32(
            /*neg_a=*/false, a, /*neg_b=*/false, b,
            /*c_mod=*/(short)0, acc, /*reuse_a=*/false, /*reuse_b=*/false);
    }

    // C/D layout: VGPR r holds (M = r, lanes 0..15) and (M = r+8, lanes 16..31)
    int col  = n_base + r16;
    int row0 = m_base + ((lane >> 4) << 3);
#pragma unroll
    for (int r = 0; r < 8; ++r) {
        out[(size_t)(row0 + r) * FD + col] = acc[r];
    }
}

// ---------------------------------------------------------------------------
// Exact correction for gaussians with rho != 0 (empty list => immediate exit).
// Adds alpha * (K_exact - K_separable) per listed gaussian.
// ---------------------------------------------------------------------------
__global__ __launch_bounds__(256) void correct_kernel(
    float* __restrict__ out,
    const float* __restrict__ mu_t, const float* __restrict__ mu_f,
    const float* __restrict__ alpha,
    const float* __restrict__ c_t, const float* __restrict__ c_f,
    const float* __restrict__ ist, const float* __restrict__ isf,
    const float* __restrict__ rho, const float* __restrict__ invden,
    const int* __restrict__ corr_list, const int* __restrict__ corr_count)
{
    int n = *corr_count;
    if (n == 0) return;
    int idx = blockIdx.x * blockDim.x + threadIdx.x;
    float tt = (float)(idx / FD);
    float ff = (float)(idx % FD);
    float acc = 0.f;
    for (int i = 0; i < n; ++i) {
        int g = corr_list[i];
        float et = tt - mu_t[g];
        float ef = ff - mu_f[g];
        float dt = et * ist[g];
        float df = ef * isf[g];
        float r  = rho[g];
        float z  = (dt * dt - 2.f * r * dt * df + df * df) * invden[g];
        float Kex  = expf(-0.5f * z);
        float Ksep = expf(-c_t[g] * et * et) * expf(-c_f[g] * ef * ef);
        acc += alpha[g] * (Kex - Ksep);
    }
    out[idx] += acc;
}

// ---------------------------------------------------------------------------
// Fully naive exact fallback (only if workspace is too small).
// ---------------------------------------------------------------------------
__global__ __launch_bounds__(256) void naive_kernel(
    const float* __restrict__ mu_t, const float* __restrict__ mu_f,
    const float* __restrict__ lst,  const float* __restrict__ lsf,
    const float* __restrict__ rrho, const float* __restrict__ ralpha,
    float* __restrict__ out)
{
    int idx = blockIdx.x * blockDim.x + threadIdx.x;
    if (idx >= TD * FD) return;
    float tt = (float)(idx / FD);
    float ff = (float)(idx % FD);
    float acc = 0.f;
    for (int g = 0; g < NG; ++g) {
        float st = expf(lst[g]);
        float sf = expf(lsf[g]);
        float r  = tanhf(rrho[g]);
        float den = 1.f - r * r + 1e-6f;
        float dt = (tt - mu_t[g]) / st;
        float df = (ff - mu_f[g]) / sf;
        float z  = (dt * dt - 2.f * r * dt * df + df * df) / den;
        acc += ralpha[g] * expf(-0.5f * z);
    }
    out[idx] = acc;
}

extern "C" void kernel_launch(void* const* d_in, const int* in_sizes, int n_in,
                              void* d_out, int out_size, void* d_ws, size_t ws_size,
                              hipStream_t stream)
{
    const float* mu_t   = (const float*)d_in[2];
    const float* mu_f   = (const float*)d_in[3];
    const float* lst    = (const float*)d_in[4];
    const float* lsf    = (const float*)d_in[5];
    const float* rrho   = (const float*)d_in[6];
    const float* ralpha = (const float*)d_in[7];
    float* out = (float*)d_out;

    // workspace layout
    size_t offA    = 0;
    size_t offB    = offA    + (size_t)TD * NG * sizeof(float);
    size_t offCt   = offB    + (size_t)FD * NG * sizeof(float);
    size_t offCf   = offCt   + (size_t)NG * sizeof(float);
    size_t offIst  = offCf   + (size_t)NG * sizeof(float);
    size_t offIsf  = offIst  + (size_t)NG * sizeof(float);
    size_t offRho  = offIsf  + (size_t)NG * sizeof(float);
    size_t offInvD = offRho  + (size_t)NG * sizeof(float);
    size_t offList = offInvD + (size_t)NG * sizeof(float);
    size_t offCnt  = offList + (size_t)NG * sizeof(int);
    size_t need    = offCnt  + sizeof(int);

    if (ws_size < need) {
        naive_kernel<<<(TD * FD + 255) / 256, 256, 0, stream>>>(
            mu_t, mu_f, lst, lsf, rrho, ralpha, out);
        return;
    }

    char* ws = (char*)d_ws;
    float* A      = (float*)(ws + offA);
    float* B      = (float*)(ws + offB);
    float* c_t    = (float*)(ws + offCt);
    float* c_f    = (float*)(ws + offCf);
    float* ist    = (float*)(ws + offIst);
    float* isf    = (float*)(ws + offIsf);
    float* rho    = (float*)(ws + offRho);
    float* invden = (float*)(ws + offInvD);
    int* corr_list  = (int*)(ws + offList);
    int* corr_count = (int*)(ws + offCnt);

    hipMemsetAsync(corr_count, 0, sizeof(int), stream);

    prep_kernel<<<(NG + 255) / 256, 256, 0, stream>>>(
        lst, lsf, rrho, c_t, c_f, ist, isf, rho, invden, corr_list, corr_count);

    build_A_kernel<<<dim3(NG / 256, TD), 256, 0, stream>>>(mu_t, c_t, ralpha, A);
    build_B_kernel<<<dim3(NG / 256, FD), 256, 0, stream>>>(mu_f, c_f, B);

    gemm_wmma_kernel<<<64, 256, 0, stream>>>(A, B, out);

    correct_kernel<<<(TD * FD) / 256, 256, 0, stream>>>(
        out, mu_t, mu_f, ralpha, c_t, c_f, ist, isf, rho, invden,
        corr_list, corr_count);
}